// HeteroGraphConvLayer_24266565222732
// MI455X (gfx1250) — compile-verified
//
#include <hip/hip_runtime.h>

typedef __attribute__((ext_vector_type(2))) float v2f;
typedef __attribute__((ext_vector_type(8))) float v8f;
typedef __attribute__((ext_vector_type(4))) unsigned int u32x4;
typedef __attribute__((ext_vector_type(8))) int i32x8;
typedef __attribute__((ext_vector_type(4))) int i32x4;

#define N_T   100000
#define N_U   20000
#define D_T   256
#define D_U   128
#define C_DIM 256
#define E_NUM 300000

// ---------------------------------------------------------------------------
// Zero-fill (grid-stride)
// ---------------------------------------------------------------------------
__global__ __launch_bounds__(256)
void fill_zero_kernel(float* __restrict__ p, int n) {
  int i = blockIdx.x * blockDim.x + threadIdx.x;
  int stride = gridDim.x * blockDim.x;
  for (; i < n; i += stride) p[i] = 0.0f;
}

// ---------------------------------------------------------------------------
// Scatter-add aggregation: agg[dst] += x_src[src] per edge.
// ---------------------------------------------------------------------------
__global__ __launch_bounds__(256)
void scatter_add_kernel(const float* __restrict__ xsrc, const int* __restrict__ ei,
                        float* __restrict__ agg, int D) {
  const int chunks = D >> 2;
  const int total  = E_NUM * chunks;
  int tid = blockIdx.x * blockDim.x + threadIdx.x;
  if (tid >= total) return;
  int e  = tid / chunks;
  int c4 = tid - e * chunks;
  int src = ei[e];
  int dst = ei[E_NUM + e];
  float4 v = ((const float4*)(xsrc + src * D))[c4];
  float* p = agg + dst * D + (c4 << 2);
  atomicAdd(p + 0, v.x);
  atomicAdd(p + 1, v.y);
  atomicAdd(p + 2, v.z);
  atomicAdd(p + 3, v.w);
}

// ---------------------------------------------------------------------------
// TDM: issue TENSOR_LOAD_TO_LDS of `rows` x 256 f32 (row-major, stride 256)
// from gsrc into LDS byte offset lds_off. D# per CDNA5 ISA:
//   g0 = { count=1 | gather=0, lds_addr, gaddr[31:0], gaddr[56:32] | type=2<<30 }
//   g1 = { data_size=2<<16, td0=256<<16, td1=rows<<16, tile0=256<<16,
//          tile1=rows, td0_stride=256, 0, 0 }
// Groups 2/3 zero (2-D tensor). Tracked by TENSORcnt.
// amdgpu-toolchain (clang-23) 6-arg builtin form.
// ---------------------------------------------------------------------------
__device__ __forceinline__ void tdm_load_rows(unsigned int lds_off,
                                              const float* gsrc, int rows) {
  unsigned long long ga = (unsigned long long)(uintptr_t)gsrc;
  u32x4 g0 = { 1u,
               lds_off,
               (unsigned int)(ga & 0xffffffffull),
               (unsigned int)((ga >> 32) & 0x01ffffffu) | (2u << 30) };
  i32x8 g1 = { (int)(2u << 16),                  // data_size = 4 bytes
               (int)(256u << 16),                // tensor_dim0 = 256
               (int)((unsigned int)rows << 16),  // tensor_dim1 = rows
               (int)(256u << 16),                // tile_dim0 = 256
               rows,                             // tile_dim1 = rows
               256,                              // tensor_dim0_stride = 256
               0, 0 };
  i32x4 z4 = { 0, 0, 0, 0 };
  i32x8 z8 = { 0, 0, 0, 0, 0, 0, 0, 0 };
  __builtin_amdgcn_tensor_load_to_lds(g0, g1, z4, z4, z8, 0);
}

// ---------------------------------------------------------------------------
// One GEMM phase: c[16] += A_strip[16 x K] @ W[K x 256], with W staged through
// LDS in CH-row panels by the TDM (wave 0 issues, double-buffered via
// s_wait_tensorcnt(1)), B read from LDS, A streamed from global once.
// ---------------------------------------------------------------------------
template<int K, int CH>
__device__ __forceinline__ void wmma_phase(const float* __restrict__ arow,
                                           const float* __restrict__ W,
                                           const float* __restrict__ sb0,
                                           const float* __restrict__ sb1,
                                           unsigned int off0, unsigned int off1,
                                           v8f c[16],
                                           int wave, int half, int mr, bool active) {
  constexpr int NCH = K / CH;
  if (wave == 0) tdm_load_rows(off0, W, CH);
  for (int ci = 0; ci < NCH; ++ci) {
    const float* sb = (ci & 1) ? sb1 : sb0;
    if (wave == 0) {
      if (ci + 1 < NCH) {
        tdm_load_rows((ci & 1) ? off0 : off1, W + (ci + 1) * CH * C_DIM, CH);
        __builtin_amdgcn_s_wait_tensorcnt(1);   // panel ci complete
      } else {
        __builtin_amdgcn_s_wait_tensorcnt(0);   // drain
      }
    }
    __syncthreads();
    if (active) {
      const float* ab = arow + ci * CH;
      __builtin_prefetch(ab + CH, 0, 0);        // global_prefetch_b8 (next A panel)
#pragma unroll
      for (int kk = 0; kk < CH; kk += 4) {
        v2f a;
        a.x = ab[kk];
        a.y = ab[kk + 1];
        const float* wk0 = sb + (kk + 2 * half) * C_DIM + mr;  // LDS reads
        const float* wk1 = wk0 + C_DIM;
#pragma unroll
        for (int nt = 0; nt < 16; ++nt) {
          v2f b;
          b.x = wk0[nt * 16];
          b.y = wk1[nt * 16];
          c[nt] = __builtin_amdgcn_wmma_f32_16x16x4_f32(
              false, a, false, b, (short)0, c[nt], false, false);
        }
      }
    }
    __syncthreads();
  }
}

// ---------------------------------------------------------------------------
// Fused GraphConv GEMM:
//   out[m,:] (+)= alpha * ( A[m,:KA] @ Wrel + brel + Xd[m,:KX] @ Wroot )
// 8 waves / block, wave = one 16-row strip, all 16 N-tiles of C=256
// (16 x v8f accumulators = 128 VGPRs under wave32). Weights staged via TDM.
// ---------------------------------------------------------------------------
template<int KA, int KX>
__global__ __launch_bounds__(256)
void gemm_graphconv_kernel(const float* __restrict__ A,  const float* __restrict__ Wrel,
                           const float* __restrict__ brel,
                           const float* __restrict__ Xd, const float* __restrict__ Wroot,
                           float* __restrict__ out, int M, float alpha, int accumulate) {
  constexpr int CH = 32;                 // K-panel rows staged per TDM op (32 KB)
  extern __shared__ float smem[];        // 2 panels, 64 KB dynamic LDS
  float* sb0 = smem;
  float* sb1 = smem + CH * C_DIM;
  const unsigned int base = (unsigned int)__builtin_amdgcn_groupstaticsize();
  const unsigned int off0 = base;
  const unsigned int off1 = base + CH * C_DIM * (unsigned int)sizeof(float);

  const int lane  = threadIdx.x & 31;
  const int wave  = threadIdx.x >> 5;
  const int mtile = blockIdx.x * 8 + wave;
  const int m0    = mtile * 16;
  const bool active = (m0 < M);          // wave-uniform; all waves keep hitting barriers
  const int half = lane >> 4;            // A/B K-half (K={0,1} vs K={2,3})
  const int mr   = lane & 15;            // A row within tile / B & D column

  v8f c[16];
#pragma unroll
  for (int nt = 0; nt < 16; ++nt) c[nt] = (v8f)(0.0f);

  const int mrow = active ? (m0 + mr) : 0;
  const float* arow = A  + mrow * KA + 2 * half;
  const float* xrow = Xd + mrow * KX + 2 * half;

  wmma_phase<KA, CH>(arow, Wrel,  sb0, sb1, off0, off1, c, wave, half, mr, active);
  wmma_phase<KX, CH>(xrow, Wroot, sb0, sb1, off0, off1, c, wave, half, mr, active);

  if (active) {
#pragma unroll
    for (int nt = 0; nt < 16; ++nt) {
      const float bias = brel[nt * 16 + mr];
#pragma unroll
      for (int r = 0; r < 8; ++r) {
        const int row = m0 + r + 8 * half;
        float v = alpha * (c[nt][r] + bias);
        float* p = out + row * C_DIM + nt * 16 + mr;
        if (accumulate) *p += v; else *p = v;
      }
    }
  }
}

// ---------------------------------------------------------------------------
// Attention: deg[dst] += 1 per edge, then att[e] = 1 / max(deg[dst[e]], 1)
// ---------------------------------------------------------------------------
__global__ __launch_bounds__(256)
void degree_kernel(const int* __restrict__ ei, float* __restrict__ deg) {
  int e = blockIdx.x * blockDim.x + threadIdx.x;
  if (e < E_NUM) atomicAdd(deg + ei[E_NUM + e], 1.0f);
}

__global__ __launch_bounds__(256)
void attention_kernel(const int* __restrict__ ei, const float* __restrict__ deg,
                      float* __restrict__ att) {
  int e = blockIdx.x * blockDim.x + threadIdx.x;
  if (e < E_NUM) {
    float d = deg[ei[E_NUM + e]];
    att[e] = 1.0f / fmaxf(d, 1.0f);
  }
}

// ---------------------------------------------------------------------------
extern "C" void kernel_launch(void* const* d_in, const int* in_sizes, int n_in,
                              void* d_out, int out_size, void* d_ws, size_t ws_size,
                              hipStream_t stream) {
  const float* x_tweet  = (const float*)d_in[0];
  const float* x_user   = (const float*)d_in[1];
  const int*   ei_rt    = (const int*)d_in[2];
  const int*   ei_rb    = (const int*)d_in[3];
  const int*   ei_po    = (const int*)d_in[4];
  const int*   ei_pb    = (const int*)d_in[5];
  const float* Wrel_rt  = (const float*)d_in[6];
  const float* brel_rt  = (const float*)d_in[7];
  const float* Wroot_rt = (const float*)d_in[8];
  const float* Wrel_rb  = (const float*)d_in[9];
  const float* brel_rb  = (const float*)d_in[10];
  const float* Wroot_rb = (const float*)d_in[11];
  const float* Wrel_po  = (const float*)d_in[12];
  const float* brel_po  = (const float*)d_in[13];
  const float* Wroot_po = (const float*)d_in[14];
  const float* Wrel_pb  = (const float*)d_in[15];
  const float* brel_pb  = (const float*)d_in[16];
  const float* Wroot_pb = (const float*)d_in[17];

  float* out    = (float*)d_out;
  float* outT   = out;                              // [N_T, C]
  float* outU   = out + (size_t)N_T * C_DIM;        // [N_U, C]
  float* att_rt = outU + (size_t)N_U * C_DIM;       // [E]
  float* att_rb = att_rt + E_NUM;
  float* att_po = att_rb + E_NUM;
  float* att_pb = att_po + E_NUM;

  // Workspace: one reusable agg buffer (max N_T x 256) + degree buffer.
  float* agg = (float*)d_ws;
  float* deg = agg + (size_t)N_T * C_DIM;

  const dim3 blk(256);
  const size_t GEMM_LDS = 2u * 32u * C_DIM * sizeof(float);   // 64 KB, 2 TDM panels
  const int egrid      = (E_NUM + 255) / 256;
  const int mblk_tweet = ((N_T / 16) + 7) / 8;      // 782 blocks, 8 waves each
  const int mblk_user  = ((N_U / 16) + 7) / 8;      // 157 blocks

  // --- relation rt: tweet -> tweet (alpha = 1/3, write) ---
  fill_zero_kernel<<<2048, blk, 0, stream>>>(agg, N_T * D_T);
  scatter_add_kernel<<<(E_NUM * (D_T / 4) + 255) / 256, blk, 0, stream>>>(x_tweet, ei_rt, agg, D_T);
  gemm_graphconv_kernel<D_T, D_T><<<mblk_tweet, blk, GEMM_LDS, stream>>>(
      agg, Wrel_rt, brel_rt, x_tweet, Wroot_rt, outT, N_T, 1.0f / 3.0f, 0);

  // --- relation rb: tweet -> tweet (accumulate) ---
  fill_zero_kernel<<<2048, blk, 0, stream>>>(agg, N_T * D_T);
  scatter_add_kernel<<<(E_NUM * (D_T / 4) + 255) / 256, blk, 0, stream>>>(x_tweet, ei_rb, agg, D_T);
  gemm_graphconv_kernel<D_T, D_T><<<mblk_tweet, blk, GEMM_LDS, stream>>>(
      agg, Wrel_rb, brel_rb, x_tweet, Wroot_rb, outT, N_T, 1.0f / 3.0f, 1);

  // --- relation po: user -> tweet (accumulate), agg width D_U ---
  fill_zero_kernel<<<2048, blk, 0, stream>>>(agg, N_T * D_U);
  scatter_add_kernel<<<(E_NUM * (D_U / 4) + 255) / 256, blk, 0, stream>>>(x_user, ei_po, agg, D_U);
  gemm_graphconv_kernel<D_U, D_T><<<mblk_tweet, blk, GEMM_LDS, stream>>>(
      agg, Wrel_po, brel_po, x_tweet, Wroot_po, outT, N_T, 1.0f / 3.0f, 1);

  // --- relation pb: tweet -> user (alpha = 1, write) ---
  fill_zero_kernel<<<2048, blk, 0, stream>>>(agg, N_U * D_T);
  scatter_add_kernel<<<(E_NUM * (D_T / 4) + 255) / 256, blk, 0, stream>>>(x_tweet, ei_pb, agg, D_T);
  gemm_graphconv_kernel<D_T, D_U><<<mblk_user, blk, GEMM_LDS, stream>>>(
      agg, Wrel_pb, brel_pb, x_user, Wroot_pb, outU, N_U, 1.0f, 0);

  // --- uniform attention per relation (reuse deg buffer sequentially) ---
  fill_zero_kernel<<<512, blk, 0, stream>>>(deg, N_T);
  degree_kernel<<<egrid, blk, 0, stream>>>(ei_rt, deg);
  attention_kernel<<<egrid, blk, 0, stream>>>(ei_rt, deg, att_rt);

  fill_zero_kernel<<<512, blk, 0, stream>>>(deg, N_T);
  degree_kernel<<<egrid, blk, 0, stream>>>(ei_rb, deg);
  attention_kernel<<<egrid, blk, 0, stream>>>(ei_rb, deg, att_rb);

  fill_zero_kernel<<<512, blk, 0, stream>>>(deg, N_T);
  degree_kernel<<<egrid, blk, 0, stream>>>(ei_po, deg);
  attention_kernel<<<egrid, blk, 0, stream>>>(ei_po, deg, att_po);

  fill_zero_kernel<<<512, blk, 0, stream>>>(deg, N_U);
  degree_kernel<<<egrid, blk, 0, stream>>>(ei_pb, deg);
  attention_kernel<<<egrid, blk, 0, stream>>>(ei_pb, deg, att_pb);
}